// _S4Block_50190987821661
// MI455X (gfx1250) — compile-verified
//
#include <hip/hip_runtime.h>
#include <hip/hip_bf16.h>
#include <math.h>

#define Hc 128
#define Nc 32
#define Bc 8
#define Lc 8192
#define TOT (Bc*Hc*Lc)

typedef __attribute__((ext_vector_type(2))) float v2f;
typedef __attribute__((ext_vector_type(8))) float v8f;

__device__ __forceinline__ float lane_bcast(float v, int t) {
  return __uint_as_float(__builtin_amdgcn_readlane(__float_as_uint(v), t));
}

// ---------------------------------------------------------------------------
// Kernel 1: per-(h,n) SSM discretization.  w = exp(dt*A),  Kc = C*(w-1)/A
// P layout (floats): [0]=wr, [4096]=wi, [8192]=K0r, [12288]=K0i, [16384]=K1r, [20480]=K1i
// ---------------------------------------------------------------------------
__global__ void k_params(const float* __restrict__ C_re, const float* __restrict__ C_im,
                         const float* __restrict__ logA, const float* __restrict__ Aim,
                         const float* __restrict__ logdt, float* __restrict__ P) {
  int i = blockIdx.x * blockDim.x + threadIdx.x;
  if (i >= Hc * Nc) return;
  int h = i >> 5;
  float Ar = -__expf(logA[i]);
  float Ai = Aim[i];
  float dt = __expf(logdt[h]);
  float dr = Ar * dt, di = Ai * dt;
  float er = __expf(dr);
  float cs, sn; __sincosf(di, &sn, &cs);
  float wr = er * cs, wi = er * sn;
  float numr = wr - 1.0f, numi = wi;
  float inv = 1.0f / (Ar * Ar + Ai * Ai);
  float qr = (numr * Ar + numi * Ai) * inv;
  float qi = (numi * Ar - numr * Ai) * inv;
  P[i]            = wr;
  P[4096 + i]     = wi;
  float c0r = C_re[i],        c0i = C_im[i];
  P[2 * 4096 + i] = c0r * qr - c0i * qi;
  P[3 * 4096 + i] = c0r * qi + c0i * qr;
  float c1r = C_re[4096 + i], c1i = C_im[4096 + i];
  P[4 * 4096 + i] = c1r * qr - c1i * qi;
  P[5 * 4096 + i] = c1r * qi + c1i * qr;
}

// ---------------------------------------------------------------------------
// Kernel 2: dual-direction diagonal-SSM scan.
// One wave32 per (direction, b, h); lane = complex mode n.
// Per 32-step chunk: x broadcast via v_readlane (no DS), contributions kept in
// 32 registers, then one LDS 32x32 transpose-reduce (8x ds_store_b128 +
// 32x conflict-free ds_load_b32) instead of a per-step shuffle butterfly.
// ---------------------------------------------------------------------------
__global__ __launch_bounds__(256) void k_scan(const float* __restrict__ x,
                                              const float* __restrict__ P,
                                              float* __restrict__ yf,
                                              float* __restrict__ yb) {
  __shared__ float tr[8][32 * 36];          // per-wave 32x32 tile, rows padded to 36
  const int lane = threadIdx.x & 31;
  const int wv   = threadIdx.x >> 5;
  const int wave = blockIdx.x * 8 + wv;     // 2048 waves
  const int dir  = wave >> 10;              // 0: causal, 1: anticausal
  const int bh   = wave & 1023;             // b*H + h
  const int h    = bh & (Hc - 1);
  const size_t base = (size_t)bh * Lc;

  const float wr = P[h * Nc + lane];
  const float wi = P[4096 + h * Nc + lane];
  const float kr = P[(dir ? 4 * 4096 : 2 * 4096) + h * Nc + lane];
  const float ki = P[(dir ? 5 * 4096 : 3 * 4096) + h * Nc + lane];

  float* rowp = &tr[wv][lane * 36];         // 144B row base: 16B aligned
  float* colp = &tr[wv][lane];
  float sr = 0.0f, si = 0.0f;
  float c[32];

  if (dir == 0) {
    for (int cc = 0; cc < Lc / 32; ++cc) {
      const float xv = x[base + cc * 32 + lane];
      if (cc + 1 < Lc / 32) __builtin_prefetch(&x[base + (cc + 1) * 32 + lane], 0, 1);
#pragma unroll
      for (int t = 0; t < 32; ++t) {
        const float xt  = lane_bcast(xv, t);
        const float nsr = fmaf(wr, sr, fmaf(-wi, si, xt));   // s = w*s + x[l]
        const float nsi = fmaf(wr, si, wi * sr);
        sr = nsr; si = nsi;
        c[t] = fmaf(kr, sr, -ki * si);                       // Re(K * s)
      }
#pragma unroll
      for (int j = 0; j < 8; ++j)
        ((float4*)rowp)[j] = make_float4(c[4*j], c[4*j+1], c[4*j+2], c[4*j+3]);
      float sum = 0.0f;
#pragma unroll
      for (int n = 0; n < 32; ++n) sum += colp[n * 36];
      yf[base + cc * 32 + lane] = 2.0f * sum;
    }
  } else {
    for (int cc = Lc / 32 - 1; cc >= 0; --cc) {
      const float xv = x[base + cc * 32 + lane];
      if (cc > 0) __builtin_prefetch(&x[base + (cc - 1) * 32 + lane], 0, 1);
#pragma unroll
      for (int t = 31; t >= 0; --t) {
        c[t] = fmaf(kr, sr, -ki * si);                       // uses r = sum_{t'>l}
        const float xt  = lane_bcast(xv, t);
        const float nsr = fmaf(wr, sr, fmaf(-wi, si, xt));   // r = x[l] + w*r
        const float nsi = fmaf(wr, si, wi * sr);
        sr = nsr; si = nsi;
      }
#pragma unroll
      for (int j = 0; j < 8; ++j)
        ((float4*)rowp)[j] = make_float4(c[4*j], c[4*j+1], c[4*j+2], c[4*j+3]);
      float sum = 0.0f;
#pragma unroll
      for (int n = 0; n < 32; ++n) sum += colp[n * 36];
      yb[base + cc * 32 + lane] = 2.0f * sum;
    }
  }
}

// ---------------------------------------------------------------------------
// Kernel 3: combine both directions + D*x skip + exact GELU (float4). -> yb
// ---------------------------------------------------------------------------
__global__ void k_combine(const float4* __restrict__ yf, float4* __restrict__ yb,
                          const float4* __restrict__ x, const float* __restrict__ D) {
  size_t i = (size_t)blockIdx.x * blockDim.x + threadIdx.x;   // over TOT/4
  int h = (int)((i >> 11) & (Hc - 1));                        // L/4 = 2^11
  const float dh = D[h];
  float4 a = yf[i], bv = yb[i], xv = x[i], g;
  float y;
  y = a.x + bv.x + dh * xv.x; g.x = 0.5f * y * (1.0f + erff(y * 0.70710678118654752f));
  y = a.y + bv.y + dh * xv.y; g.y = 0.5f * y * (1.0f + erff(y * 0.70710678118654752f));
  y = a.z + bv.z + dh * xv.z; g.z = 0.5f * y * (1.0f + erff(y * 0.70710678118654752f));
  y = a.w + bv.w + dh * xv.w; g.w = 0.5f * y * (1.0f + erff(y * 0.70710678118654752f));
  yb[i] = g;
}

// ---------------------------------------------------------------------------
// Kernel 4: pointwise conv  out[b,o,l] = sum_h W[o,h]*act[b,h,l]  via
// V_WMMA_F32_16X16X4_F32.  One wave per 16x16 output tile, K=128 in 32 steps.
// ---------------------------------------------------------------------------
__global__ __launch_bounds__(256) void k_gemm(const float* __restrict__ act,
                                              const float* __restrict__ W,
                                              float* __restrict__ out) {
  const int blk   = blockIdx.x;        // 8(b) * 8(mtile) * 64(ngrp)
  const int b     = blk >> 9;
  const int rem   = blk & 511;
  const int mtile = rem >> 6;
  const int ngrp  = rem & 63;

  __shared__ float Wt[16 * 129];       // +1 pad per row -> conflict-free column reads
  for (int i = threadIdx.x; i < 16 * 128; i += 256) {
    int m = i >> 7, k = i & 127;
    Wt[m * 129 + k] = W[(mtile * 16 + m) * Hc + k];
  }
  __syncthreads();

  const int waveid = threadIdx.x >> 5;
  const int lane   = threadIdx.x & 31;
  const int half   = lane >> 4;        // 0: lanes 0-15, 1: lanes 16-31
  const int lm     = lane & 15;
  const int l0     = (ngrp * 8 + waveid) * 16;

  const float* ap = act + ((size_t)b * Hc) * Lc + l0 + lm;
  v8f acc = {0.f, 0.f, 0.f, 0.f, 0.f, 0.f, 0.f, 0.f};

#pragma unroll 4
  for (int k0 = 0; k0 < Hc; k0 += 4) {
    const int krow = k0 + 2 * half;    // ISA A/B layout: K 0,1 | 2,3 split by lane half
    v2f a, bfrag;
    a.x = Wt[lm * 129 + krow];
    a.y = Wt[lm * 129 + krow + 1];
    bfrag.x = ap[(size_t)krow * Lc];
    bfrag.y = ap[(size_t)(krow + 1) * Lc];
    acc = __builtin_amdgcn_wmma_f32_16x16x4_f32(false, a, false, bfrag,
                                                (short)0, acc, false, false);
  }

  float* yp = out + ((size_t)b * Hc + mtile * 16) * Lc + l0 + lm;
#pragma unroll
  for (int r = 0; r < 8; ++r)
    yp[(size_t)(r + 8 * half) * Lc] = acc[r];   // D rows r / r+8 per lane half
}

// ---------------------------------------------------------------------------
// Kernel 5: BatchNorm batch statistics per output channel (one block each).
// ---------------------------------------------------------------------------
__global__ void k_bnstats(const float* __restrict__ y, const float* __restrict__ gamma,
                          float* __restrict__ mean, float* __restrict__ scale) {
  const int o = blockIdx.x, tid = threadIdx.x;
  float s = 0.0f, s2 = 0.0f;
  const float4* y4 = (const float4*)y;
  for (int i = tid; i < Bc * Lc / 4; i += 256) {
    int b = i >> 11, l4 = i & 2047;                  // L/4 = 2048 float4 per row
    float4 v = y4[((size_t)b * Hc + o) * (Lc / 4) + l4];
    s  += v.x + v.y + v.z + v.w;
    s2 += v.x * v.x + v.y * v.y + v.z * v.z + v.w * v.w;
  }
  __shared__ float sh[256], sh2[256];
  sh[tid] = s; sh2[tid] = s2;
  __syncthreads();
  for (int off = 128; off > 0; off >>= 1) {
    if (tid < off) { sh[tid] += sh[tid + off]; sh2[tid] += sh2[tid + off]; }
    __syncthreads();
  }
  if (tid == 0) {
    float m   = sh[0] * (1.0f / 65536.0f);
    float var = sh2[0] * (1.0f / 65536.0f) - m * m;
    mean[o]  = m;
    scale[o] = rsqrtf(var + 1e-5f) * gamma[o];
  }
}

// ---------------------------------------------------------------------------
// Kernel 6: BN apply + ELU + residual, in place on d_out (float4).
// ---------------------------------------------------------------------------
__global__ void k_final(float4* __restrict__ y, const float4* __restrict__ x,
                        const float* __restrict__ mean, const float* __restrict__ scale,
                        const float* __restrict__ beta) {
  size_t i = (size_t)blockIdx.x * blockDim.x + threadIdx.x;   // over TOT/4
  int h = (int)((i >> 11) & (Hc - 1));
  const float m = mean[h], sc = scale[h], bt = beta[h];
  float4 v = y[i], xv = x[i], r;
  float t;
  t = (v.x - m) * sc + bt; r.x = (t > 0.0f ? t : expm1f(t)) + xv.x;
  t = (v.y - m) * sc + bt; r.y = (t > 0.0f ? t : expm1f(t)) + xv.y;
  t = (v.z - m) * sc + bt; r.z = (t > 0.0f ? t : expm1f(t)) + xv.z;
  t = (v.w - m) * sc + bt; r.w = (t > 0.0f ? t : expm1f(t)) + xv.w;
  y[i] = r;
}

// ---------------------------------------------------------------------------
extern "C" void kernel_launch(void* const* d_in, const int* in_sizes, int n_in,
                              void* d_out, int out_size, void* d_ws, size_t ws_size,
                              hipStream_t stream) {
  const float* x     = (const float*)d_in[0];
  const float* C_re  = (const float*)d_in[1];
  const float* C_im  = (const float*)d_in[2];
  const float* logA  = (const float*)d_in[3];
  const float* Aim   = (const float*)d_in[4];
  const float* logdt = (const float*)d_in[5];
  const float* D     = (const float*)d_in[6];
  const float* W     = (const float*)d_in[7];
  const float* gamma = (const float*)d_in[8];
  const float* beta  = (const float*)d_in[9];

  float* out = (float*)d_out;
  float* wsf = (float*)d_ws;
  float* P     = wsf;                 // 6 * 4096 floats of SSM parameters
  float* MEAN  = wsf + 6 * 4096;      // 128 floats
  float* SCALE = MEAN + 128;          // 128 floats
  float* WSB   = wsf + (1 << 18);     // 1 MiB offset: B*H*L float buffer (33.5 MB)

  // 1. discretize SSM
  k_params<<<16, 256, 0, stream>>>(C_re, C_im, logA, Aim, logdt, P);
  // 2. dual scans: causal -> d_out (scratch), anticausal -> WSB
  k_scan<<<256, 256, 0, stream>>>(x, P, out, WSB);
  // 3. combine + skip + GELU -> WSB (activation)
  k_combine<<<TOT / 4 / 256, 256, 0, stream>>>((const float4*)out, (float4*)WSB,
                                               (const float4*)x, D);
  // 4. 128x128 pointwise conv via f32 WMMA -> d_out
  k_gemm<<<8 * 8 * 64, 256, 0, stream>>>(WSB, W, out);
  // 5. BN statistics
  k_bnstats<<<Hc, 256, 0, stream>>>(out, gamma, MEAN, SCALE);
  // 6. BN + ELU + residual (in place)
  k_final<<<TOT / 4 / 256, 256, 0, stream>>>((float4*)out, (const float4*)x,
                                             MEAN, SCALE, beta);
}